// SpectralGroupAttention_257698038026
// MI455X (gfx1250) — compile-verified
//
#include <hip/hip_runtime.h>
#include <hip/hip_bf16.h>
#include <math.h>

// ---------------------------------------------------------------------------
// SpectralGroupAttention fused pipeline for MI455X (gfx1250, wave32, WMMA).
// GEMMs use V_WMMA_F32_16X16X4_F32 (full f32 precision). Each wave owns a
// 16(M) x 32(N) tile: one A-fragment feeds two WMMAs per k-step, halving
// activation traffic vs a 16x16 tile (the f32 WMMA loop is load-issue bound:
// 16x16 tile = 512 B loaded per 2048-FLOP wmma). The S6 selective scan is
// fused into registers: one wave per (b,d) chain, 4 states per lane,
// butterfly reduction for the C.h contraction -- never materializes the
// [B,L,D,N] tensors (saves ~5 GB of HBM traffic; all intermediates ~40 MB,
// mostly resident in the 192 MB L2).
// ---------------------------------------------------------------------------

typedef __attribute__((ext_vector_type(2))) float v2f;
typedef __attribute__((ext_vector_type(8))) float v8f;

#define B_SZ   256
#define BAND   189
#define GLEN   20
#define DMODEL 128
#define DI     256
#define NSTATE 128
#define LSEQ   34
#define MROWS  (B_SZ * LSEQ)   // 8704, divisible by 16

enum { EPI_NONE = 0, EPI_LEAKY = 1, EPI_SILU = 2, EPI_GATE = 3, EPI_SOFTPLUS = 4 };

__device__ __forceinline__ float silu_f(float v) { return v / (1.f + expf(-v)); }

// ---------------------------------------------------------------------------
// K1: grouped conv1d(1->128, k=20, stride=5) + LeakyReLU + RMSNorm
// one block per (b,l) row, 128 threads = one channel each
// ---------------------------------------------------------------------------
__global__ __launch_bounds__(128)
void conv_rms_kernel(const float* __restrict__ x, const float* __restrict__ cw,
                     const float* __restrict__ cb, const float* __restrict__ nw,
                     float* __restrict__ xn)
{
    int m = blockIdx.x;                 // b*LSEQ + l
    int b = m / LSEQ, l = m - b * LSEQ;
    int c = threadIdx.x;
    const float* xp = x + (size_t)b * BAND + l * 5;
    float acc = cb[c];
#pragma unroll
    for (int k = 0; k < GLEN; ++k) acc = fmaf(xp[k], cw[c * GLEN + k], acc);
    acc = acc >= 0.f ? acc : 0.01f * acc;          // LeakyReLU
    __shared__ float red[DMODEL];
    red[c] = acc * acc;
    __syncthreads();
    for (int s = DMODEL / 2; s > 0; s >>= 1) {
        if (c < s) red[c] += red[c + s];
        __syncthreads();
    }
    float inv = rsqrtf(red[0] * (1.f / DMODEL) + 1e-5f);
    xn[(size_t)m * DMODEL + c] = acc * inv * nw[c];
}

// ---------------------------------------------------------------------------
// WMMA GEMM: Out[M,N] = epi(X[M,K] @ W[N,K]^T + bias[N])  (PyTorch Linear)
// One 16x32 output tile per wave (two v8f accumulators sharing one A frag),
// K stepped by 4 via v_wmma_f32_16x16x4_f32. N must be a multiple of 32.
// EPI_GATE: out = silu(acc+bias) * silu(gate[m,n])  (gate may alias Out;
// each element is read before its in-thread write, so in-place is safe).
// ---------------------------------------------------------------------------
template <int EPI>
__global__ __launch_bounds__(256)
void gemm16_kernel(const float* __restrict__ X, const float* __restrict__ W,
                   const float* __restrict__ bias, const float* gate,
                   float* Out, int M, int N, int K)
{
    const int lane = threadIdx.x & 31;
    const int wav  = threadIdx.x >> 5;
    const int tile = blockIdx.x * 8 + wav;
    const int tn_cnt = N >> 5;                  // 32-wide N tiles
    const int tm = tile / tn_cnt;
    const int tn = tile - tm * tn_cnt;
    if ((tm << 4) >= M) return;                 // whole-wave uniform exit
    const int m0 = tm << 4, n0 = tn << 5;
    const int ml = lane & 15;                   // row index within 16-slot
    const int kh = lane >> 4;                   // K-half select (0: k0..1, 1: k2..3)

    const float* xrow  = X + (size_t)(m0 + ml) * K + 2 * kh;
    const float* wrow0 = W + (size_t)(n0 + ml) * K + 2 * kh;
    const float* wrow1 = W + (size_t)(n0 + 16 + ml) * K + 2 * kh;

    v8f acc0 = {}, acc1 = {};
#pragma unroll 4
    for (int k = 0; k < K; k += 4) {
        v2f a  = *(const v2f*)(xrow + k);       // A[m, k+2kh : k+2kh+1]
        v2f b0 = *(const v2f*)(wrow0 + k);      // B[k.., n0+ml]    = W[n, k..]
        v2f b1 = *(const v2f*)(wrow1 + k);      // B[k.., n0+16+ml]
        acc0 = __builtin_amdgcn_wmma_f32_16x16x4_f32(
            false, a, false, b0, (short)0, acc0, false, false);
        acc1 = __builtin_amdgcn_wmma_f32_16x16x4_f32(
            false, a, false, b1, (short)0, acc1, false, false);
    }

    const float bn0 = bias ? bias[n0 + ml] : 0.f;
    const float bn1 = bias ? bias[n0 + 16 + ml] : 0.f;
#pragma unroll
    for (int j = 0; j < 8; ++j) {
        int m = m0 + j + 8 * kh;                // C/D layout: lanes 16-31 -> M+8
#pragma unroll
        for (int half = 0; half < 2; ++half) {
            int n = n0 + 16 * half + ml;
            float v = (half ? acc1[j] + bn1 : acc0[j] + bn0);
            if constexpr (EPI == EPI_LEAKY) {
                v = v >= 0.f ? v : 0.01f * v;
            } else if constexpr (EPI == EPI_SILU) {
                v = silu_f(v);
            } else if constexpr (EPI == EPI_SOFTPLUS) {
                v = (v > 20.f) ? v : log1pf(expf(v));
            } else if constexpr (EPI == EPI_GATE) {
                float g = gate[(size_t)m * N + n];     // read BEFORE in-place write
                v = silu_f(v) * silu_f(g);
            }
            Out[(size_t)m * N + n] = v;
        }
    }
}

// ---------------------------------------------------------------------------
// K3: Conv1d over the L axis (channels = L = 34, k = 3, pad = 1) + SiLU
// one block per batch, LDS-staged [34 x 258] activation slab
// ---------------------------------------------------------------------------
__global__ __launch_bounds__(256)
void seqconv_kernel(const float* __restrict__ xp, const float* __restrict__ w,
                    const float* __restrict__ bias, float* __restrict__ xc)
{
    __shared__ float S[LSEQ][DI + 2];           // S[l][c+1] = xp[b,l,c], zero edges
    int b = blockIdx.x;
    int t = threadIdx.x;
    for (int i = t; i < LSEQ * DI; i += 256) {
        int l = i >> 8, c = i & (DI - 1);
        S[l][c + 1] = xp[((size_t)b * LSEQ + l) * DI + c];
    }
    if (t < LSEQ) { S[t][0] = 0.f; S[t][DI + 1] = 0.f; }
    __syncthreads();
    int c = t;                                   // feature index 0..255
    for (int lo = 0; lo < LSEQ; ++lo) {
        float acc = bias[lo];
        const float* wrow = w + lo * LSEQ * 3;
#pragma unroll 2
        for (int li = 0; li < LSEQ; ++li) {
            acc = fmaf(wrow[li * 3 + 0], S[li][c],     acc);
            acc = fmaf(wrow[li * 3 + 1], S[li][c + 1], acc);
            acc = fmaf(wrow[li * 3 + 2], S[li][c + 2], acc);
        }
        xc[((size_t)b * LSEQ + lo) * DI + c] = silu_f(acc);
    }
}

// ---------------------------------------------------------------------------
// K8: S6 selective scan, fully fused.
// One wave per (b,d) recurrence chain; lane holds 4 states n = 4*lane..4*lane+3.
// h <- exp(delta*A)*h + delta*x*B ; out[b,l,d] = sum_n C[b,l,n]*h  (butterfly).
// ---------------------------------------------------------------------------
__global__ __launch_bounds__(256)
void ssm_scan_kernel(const float* __restrict__ delta, const float* __restrict__ xin,
                     const float* __restrict__ Bm, const float* __restrict__ Cm,
                     const float* __restrict__ A, float* __restrict__ xs)
{
    int gw   = (blockIdx.x * 256 + threadIdx.x) >> 5;   // global wave id
    int lane = threadIdx.x & 31;
    int d = gw & (DI - 1);
    int b = gw >> 8;
    int n = lane << 2;
    float4 a4 = *(const float4*)(A + (size_t)d * NSTATE + n);
    float h0 = 0.f, h1 = 0.f, h2 = 0.f, h3 = 0.f;
    for (int l = 0; l < LSEQ; ++l) {
        size_t m = (size_t)b * LSEQ + l;
        float dl = delta[m * DI + d];
        float xv = xin[m * DI + d];
        float4 Bv = *(const float4*)(Bm + m * NSTATE + n);
        float4 Cv = *(const float4*)(Cm + m * NSTATE + n);
        float dx = dl * xv;
        h0 = fmaf(expf(dl * a4.x), h0, dx * Bv.x);
        h1 = fmaf(expf(dl * a4.y), h1, dx * Bv.y);
        h2 = fmaf(expf(dl * a4.z), h2, dx * Bv.z);
        h3 = fmaf(expf(dl * a4.w), h3, dx * Bv.w);
        float p = Cv.x * h0 + Cv.y * h1 + Cv.z * h2 + Cv.w * h3;
#pragma unroll
        for (int off = 16; off > 0; off >>= 1) p += __shfl_xor(p, off, 32);
        if (lane == 0) xs[m * DI + d] = p;
    }
}

// ---------------------------------------------------------------------------
static void launch_gemm(hipStream_t s, int epi, const float* X, const float* W,
                        const float* bias, const float* gate, float* out,
                        int M, int N, int K)
{
    int tiles = (M >> 4) * (N >> 5);            // 16x32 tiles per wave
    dim3 grid((tiles + 7) >> 3), block(256);
    switch (epi) {
    case EPI_NONE:     gemm16_kernel<EPI_NONE><<<grid, block, 0, s>>>(X, W, bias, gate, out, M, N, K); break;
    case EPI_LEAKY:    gemm16_kernel<EPI_LEAKY><<<grid, block, 0, s>>>(X, W, bias, gate, out, M, N, K); break;
    case EPI_SILU:     gemm16_kernel<EPI_SILU><<<grid, block, 0, s>>>(X, W, bias, gate, out, M, N, K); break;
    case EPI_GATE:     gemm16_kernel<EPI_GATE><<<grid, block, 0, s>>>(X, W, bias, gate, out, M, N, K); break;
    case EPI_SOFTPLUS: gemm16_kernel<EPI_SOFTPLUS><<<grid, block, 0, s>>>(X, W, bias, gate, out, M, N, K); break;
    }
}

extern "C" void kernel_launch(void* const* d_in, const int* in_sizes, int n_in,
                              void* d_out, int out_size, void* d_ws, size_t ws_size,
                              hipStream_t stream)
{
    (void)in_sizes; (void)n_in; (void)out_size; (void)ws_size;

    const float* x         = (const float*)d_in[0];
    const float* conv_w    = (const float*)d_in[1];
    const float* conv_b    = (const float*)d_in[2];
    const float* norm_w    = (const float*)d_in[3];
    const float* inp_w     = (const float*)d_in[4];
    const float* inp_b     = (const float*)d_in[5];
    const float* seqconv_w = (const float*)d_in[6];
    const float* seqconv_b = (const float*)d_in[7];
    const float* convlin_w = (const float*)d_in[8];
    const float* convlin_b = (const float*)d_in[9];
    const float* fc1_w     = (const float*)d_in[10];
    const float* fc1_b     = (const float*)d_in[11];
    const float* fc2_w     = (const float*)d_in[12];
    const float* fc2_b     = (const float*)d_in[13];
    const float* fc3_w     = (const float*)d_in[14];
    const float* fc3_b     = (const float*)d_in[15];
    const float* A_mat     = (const float*)d_in[16];
    const float* D_w       = (const float*)d_in[17];
    const float* D_b       = (const float*)d_in[18];
    const float* out_w     = (const float*)d_in[19];
    const float* out_b     = (const float*)d_in[20];
    const float* fnn1_w    = (const float*)d_in[21];
    const float* fnn1_b    = (const float*)d_in[22];
    const float* fnn2_w    = (const float*)d_in[23];
    const float* fnn2_b    = (const float*)d_in[24];

    // Workspace carving (~40 MB of f32), with lifetime-based reuse.
    float* ws  = (float*)d_ws;
    float* xn  = ws;                                  // [8704,128]  (later: y2)
    float* xp  = xn + (size_t)MROWS * DMODEL;         // [8704,256]  (later: xc2)
    float* xc  = xp + (size_t)MROWS * DI;             // [8704,256]  (later: delta)
    float* Bm  = xc + (size_t)MROWS * DI;             // [8704,128]  (later: f1)
    float* Cm  = Bm + (size_t)MROWS * NSTATE;         // [8704,128]
    float* xs  = Cm + (size_t)MROWS * NSTATE;         // [8704,256]  (later: y, in place)
    float* xc2 = xp;
    float* dlt = xc;
    float* y   = xs;
    float* y2  = xn;
    float* f1  = Bm;

    // 1. conv + leaky + rmsnorm -> xn
    conv_rms_kernel<<<MROWS, 128, 0, stream>>>(x, conv_w, conv_b, norm_w, xn);
    // 2. xp = xn @ inp_w^T + inp_b
    launch_gemm(stream, EPI_NONE, xn, inp_w, inp_b, nullptr, xp, MROWS, DI, DMODEL);
    // 3. seq conv over L + silu -> xc
    seqconv_kernel<<<B_SZ, 256, 0, stream>>>(xp, seqconv_w, seqconv_b, xc);
    // 4. xc2 = xc @ convlin_w^T + convlin_b   (overwrites dead xp)
    launch_gemm(stream, EPI_NONE, xc, convlin_w, convlin_b, nullptr, xc2, MROWS, DI, DI);
    // 5. delta = softplus(xc2 @ fc1^T + b)    (overwrites dead xc)
    launch_gemm(stream, EPI_SOFTPLUS, xc2, fc1_w, fc1_b, nullptr, dlt, MROWS, DI, DI);
    // 6/7. Bm, Cm
    launch_gemm(stream, EPI_NONE, xc2, fc2_w, fc2_b, nullptr, Bm, MROWS, NSTATE, DI);
    launch_gemm(stream, EPI_NONE, xc2, fc3_w, fc3_b, nullptr, Cm, MROWS, NSTATE, DI);
    // 8. fused selective scan -> xs   (256*256 waves, 8 waves/block)
    ssm_scan_kernel<<<(B_SZ * DI * 32) / 256, 256, 0, stream>>>(dlt, xc2, Bm, Cm, A_mat, xs);
    // 9. y = silu(xs) * silu(xn @ D_w^T + D_b)   (in place over xs)
    launch_gemm(stream, EPI_GATE, xn, D_w, D_b, xs, y, MROWS, DI, DMODEL);
    // 10. y2 = y @ out_w^T + out_b   (overwrites dead xn)
    launch_gemm(stream, EPI_NONE, y, out_w, out_b, nullptr, y2, MROWS, DMODEL, DI);
    // 11. FNN head: y2 viewed as [256, 4352]
    launch_gemm(stream, EPI_LEAKY, y2, fnn1_w, fnn1_b, nullptr, f1, B_SZ, 64, LSEQ * DMODEL);
    launch_gemm(stream, EPI_NONE, f1, fnn2_w, fnn2_b, nullptr, (float*)d_out, B_SZ, 32, 64);
}